// FraudGCN_51814485459563
// MI455X (gfx1250) — compile-verified
//
#include <hip/hip_runtime.h>
#include <hip/hip_bf16.h>

// FraudGCN forward for MI455X (gfx1250, wave32).
// Dense GEMMs -> V_WMMA_F32_16X16X4_F32, 16x64 output per wave (4 accumulators,
//                A-fragment register reuse), epilogue mode templated so no
//                runtime branches / accumulator copies.
// Edge aggregation -> wave-per-edge, lane-per-channel coalesced atomics.

typedef float v2f __attribute__((ext_vector_type(2)));
typedef float v8f __attribute__((ext_vector_type(8)));

#define TPB 256

// ---------------- utility ----------------
__global__ void k_fill(float* __restrict__ p, float v, long n) {
  long i = (long)blockIdx.x * blockDim.x + threadIdx.x;
  if (i < n) p[i] = v;
}

__device__ inline void atomicMaxF(float* addr, float val) {
  unsigned int* ua = (unsigned int*)addr;
  unsigned int old = __float_as_uint(*addr);
  while (true) {
    float cur = __uint_as_float(old);
    if (!(val > cur)) break;
    unsigned int prev = atomicCAS(ua, old, __float_as_uint(val));
    if (prev == old) break;
    old = prev;
  }
}

// ---------------- WMMA GEMM ----------------
// C[M,Nc] = op(A[M,K] @ B[K,Nc]); one wave per 16x64 output strip
// (4 x 16x16 WMMA tiles sharing the A fragment). grid=(M/16, Nc/64), block=32.
// MODE: 0 = overwrite, 1 = accumulate into C, 2 = overwrite + relu
// Fragment layouts per CDNA5 ISA 7.12.2:
//   A 16x4 f32 : lanes 0-15 -> M=lane, VGPR0=K0,VGPR1=K1 ; lanes 16-31 -> K2,K3
//   B 4x16 f32 : mirrored with N=lane&15
//   C 16x16    : VGPR r -> M=r (lanes 0-15) / M=8+r (lanes 16-31), N=lane&15
template <int MODE, bool HAS_BIAS>
__global__ void k_wmma_gemm(const float* __restrict__ A, int lda,
                            const float* __restrict__ B, int ldb,
                            const float* __restrict__ bias,
                            float* __restrict__ C, int ldc, int K) {
  const int lane = threadIdx.x & 31;
  const int rowBase = blockIdx.x * 16;
  const int colBase = blockIdx.y * 64;
  const int m  = lane & 15;
  const int kg = (lane >> 4) << 1;  // 0 or 2
  const int orow = rowBase + ((lane >> 4) << 3);

  v8f acc[4];
#pragma unroll
  for (int g = 0; g < 4; ++g)
#pragma unroll
    for (int r = 0; r < 8; ++r)
      acc[g][r] = (MODE == 1) ? C[(long)(orow + r) * ldc + colBase + g * 16 + m] : 0.0f;

#if __has_builtin(__builtin_amdgcn_wmma_f32_16x16x4_f32)
  const float* Arow = A + (long)(rowBase + m) * lda;
  const float* Bb   = B + colBase + m;
  for (int k = 0; k < K; k += 4) {
    v2f a;
    a.x = Arow[k + kg];
    a.y = Arow[k + kg + 1];
#pragma unroll
    for (int g = 0; g < 4; ++g) {
      v2f b;
      b.x = Bb[(long)(k + kg) * ldb + g * 16];
      b.y = Bb[(long)(k + kg + 1) * ldb + g * 16];
      acc[g] = __builtin_amdgcn_wmma_f32_16x16x4_f32(false, a, false, b,
                                                     (short)0, acc[g], false, false);
    }
  }
#else
  for (int k = 0; k < K; ++k)
#pragma unroll
    for (int g = 0; g < 4; ++g) {
      float bv = B[(long)k * ldb + colBase + g * 16 + m];
#pragma unroll
      for (int r = 0; r < 8; ++r)
        acc[g][r] += A[(long)(orow + r) * lda + k] * bv;
    }
#endif

#pragma unroll
  for (int g = 0; g < 4; ++g) {
    const int ocol = colBase + g * 16 + m;
    const float bv = HAS_BIAS ? bias[ocol] : 0.0f;
#pragma unroll
    for (int r = 0; r < 8; ++r) {
      float v = acc[g][r] + bv;
      if (MODE == 2) v = fmaxf(v, 0.0f);
      C[(long)(orow + r) * ldc + ocol] = v;
    }
  }
}

// ---------------- GAT attention pieces ----------------
// wave per (node,head): es/ed[n*H+h] = sum_c h[n,h*C+c] * a_{src,dst}[h*C+c]
__global__ void k_gat_edot(const float* __restrict__ h,
                           const float* __restrict__ asrc,
                           const float* __restrict__ adst,
                           float* __restrict__ es, float* __restrict__ ed,
                           int n, int H, int C) {
  long wid = ((long)blockIdx.x * blockDim.x + threadIdx.x) >> 5;
  int lane = threadIdx.x & 31;
  if (wid >= (long)n * H) return;
  int node = (int)(wid / H), hh = (int)(wid - (long)node * H);
  const float* hp = h + (long)node * H * C + hh * C;
  const float* as = asrc + hh * C;
  const float* ad = adst + hh * C;
  float s0 = 0.f, s1 = 0.f;
  for (int c = lane; c < C; c += 32) {
    float v = hp[c];
    s0 += v * as[c];
    s1 += v * ad[c];
  }
#pragma unroll
  for (int off = 16; off > 0; off >>= 1) {
    s0 += __shfl_xor(s0, off, 32);
    s1 += __shfl_xor(s1, off, 32);
  }
  if (lane == 0) { es[wid] = s0; ed[wid] = s1; }
}

// segment max of leaky_relu(es[src]+ed[dst]) over incoming edges (+self loops)
__global__ void k_gat_max(const int* __restrict__ src, const int* __restrict__ dst,
                          int E, int n,
                          const float* __restrict__ es, const float* __restrict__ ed,
                          float* __restrict__ mx, int H) {
  long t = (long)blockIdx.x * blockDim.x + threadIdx.x;
  long EL = (long)E + n;
  if (t >= EL) return;
  int s = (t < E) ? src[t] : (int)(t - E);
  int d = (t < E) ? dst[t] : (int)(t - E);
  for (int h = 0; h < H; ++h) {
    float v = es[s * H + h] + ed[d * H + h];
    v = (v > 0.f) ? v : 0.2f * v;
    atomicMaxF(&mx[d * H + h], v);
  }
}

__global__ void k_gat_expsum(const int* __restrict__ src, const int* __restrict__ dst,
                             int E, int n,
                             const float* __restrict__ es, const float* __restrict__ ed,
                             const float* __restrict__ mx,
                             float* __restrict__ exbuf, float* __restrict__ den, int H) {
  long t = (long)blockIdx.x * blockDim.x + threadIdx.x;
  long EL = (long)E + n;
  if (t >= EL) return;
  int s = (t < E) ? src[t] : (int)(t - E);
  int d = (t < E) ? dst[t] : (int)(t - E);
  for (int h = 0; h < H; ++h) {
    float v = es[s * H + h] + ed[d * H + h];
    v = (v > 0.f) ? v : 0.2f * v;
    float mm = mx[d * H + h];
    if (mm < -1e29f) mm = 0.f;  // isfinite guard of reference
    float ex = expf(v - mm);
    exbuf[t * H + h] = ex;
    atomicAdd(&den[d * H + h], ex);
  }
}

// wave per (edge,head), lanes cover channels: coalesced 128B atomic bursts
__global__ void k_gat_agg(const int* __restrict__ src, const int* __restrict__ dst,
                          int E, int n,
                          const float* __restrict__ hfeat,
                          const float* __restrict__ exbuf,
                          const float* __restrict__ den,
                          float* __restrict__ out,
                          int H, int C, int ldout) {
  long wid = ((long)blockIdx.x * blockDim.x + threadIdx.x) >> 5;
  int lane = threadIdx.x & 31;
  long EL = (long)E + n;
  if (wid >= EL * H) return;
  long t = wid / H;
  int h = (int)(wid - t * H);
  int s = (t < E) ? src[t] : (int)(t - E);
  int d = (t < E) ? dst[t] : (int)(t - E);
  float alpha = exbuf[wid] / den[d * H + h];
  const float* hp = hfeat + (long)s * H * C + h * C;
  float* op = out + (long)d * ldout + h * C;
  for (int c = lane; c < C; c += 32) atomicAdd(&op[c], alpha * hp[c]);
}

// ---------------- SAGE mean aggregation ----------------
// wave per edge, lanes cover channels
__global__ void k_sage_agg(const float* __restrict__ x,
                           const int* __restrict__ src, const int* __restrict__ dst,
                           int E, float* __restrict__ agg, float* __restrict__ cnt, int F) {
  long wid = ((long)blockIdx.x * blockDim.x + threadIdx.x) >> 5;
  int lane = threadIdx.x & 31;
  if (wid >= E) return;
  int s = src[wid], d = dst[wid];
  if (lane == 0) atomicAdd(&cnt[d], 1.0f);
  const float* xp = x + (long)s * F;
  float* ap = agg + (long)d * F;
  for (int c = lane; c < F; c += 32) atomicAdd(&ap[c], xp[c]);
}

__global__ void k_sage_div(float* __restrict__ agg, const float* __restrict__ cnt,
                           int n, int F) {
  long i = (long)blockIdx.x * blockDim.x + threadIdx.x;
  if (i >= (long)n * F) return;
  agg[i] /= fmaxf(cnt[i / F], 1.0f);
}

// ---------------- BatchNorm (biased var), two-phase coalesced ----------------
// phase 1: blockDim.x == cols, blocks stride rows; fully coalesced row reads
__global__ void k_bn_part(const float* __restrict__ x, int n, int cols,
                          float* __restrict__ sum, float* __restrict__ sumsq) {
  int c = threadIdx.x;
  float s = 0.f, q = 0.f;
  for (int r = blockIdx.x; r < n; r += gridDim.x) {
    float v = x[(long)r * cols + c];
    s += v; q += v * v;
  }
  atomicAdd(&sum[c], s);
  atomicAdd(&sumsq[c], q);
}

__global__ void k_bn_finish(const float* __restrict__ sum, const float* __restrict__ sumsq,
                            int n, int cols, float* __restrict__ mu, float* __restrict__ rs) {
  int c = blockIdx.x * blockDim.x + threadIdx.x;
  if (c >= cols) return;
  float m = sum[c] / n;
  float var = sumsq[c] / n - m * m;
  mu[c] = m;
  rs[c] = rsqrtf(var + 1e-5f);
}

// act: 0=none 1=relu 2=elu ; in-place
__global__ void k_bn_apply(float* __restrict__ x, long total, int cols,
                           const float* __restrict__ mu, const float* __restrict__ rs,
                           const float* __restrict__ g, const float* __restrict__ b,
                           int act) {
  long i = (long)blockIdx.x * blockDim.x + threadIdx.x;
  if (i >= total) return;
  int c = (int)(i % cols);
  float v = (x[i] - mu[c]) * rs[c] * g[c] + b[c];
  if (act == 1) v = fmaxf(v, 0.f);
  else if (act == 2) v = (v > 0.f) ? v : expm1f(v);
  x[i] = v;
}

__global__ void k_bias_add(float* __restrict__ x, const float* __restrict__ b,
                           int n, int cols, int ld) {
  long i = (long)blockIdx.x * blockDim.x + threadIdx.x;
  if (i >= (long)n * cols) return;
  long r = i / cols; int c = (int)(i - r * cols);
  x[r * ld + c] += b[c];
}

// final logits: out[n,2] = hf[n,:64] @ Wf2[64,2] + bf2
__global__ void k_final(const float* __restrict__ hf, const float* __restrict__ Wf2,
                        const float* __restrict__ bf2, float* __restrict__ out, int n) {
  int i = blockIdx.x * blockDim.x + threadIdx.x;
  if (i >= n) return;
  const float* r = hf + (long)i * 64;
  float a0 = bf2[0], a1 = bf2[1];
  for (int c = 0; c < 64; ++c) {
    float v = r[c];
    a0 += v * Wf2[c * 2 + 0];
    a1 += v * Wf2[c * 2 + 1];
  }
  out[(long)i * 2 + 0] = a0;
  out[(long)i * 2 + 1] = a1;
}

// ---------------- launcher ----------------
static inline int cdiv(long a, long b) { return (int)((a + b - 1) / b); }

extern "C" void kernel_launch(void* const* d_in, const int* in_sizes, int n_in,
                              void* d_out, int out_size, void* d_ws, size_t ws_size,
                              hipStream_t stream) {
  const float* x     = (const float*)d_in[0];
  const int*   ei    = (const int*)  d_in[1];
  const float* W1g   = (const float*)d_in[2];
  const float* asrc1 = (const float*)d_in[3];
  const float* adst1 = (const float*)d_in[4];
  // d_in[5] = b1g: shift-invariant through BN -> mathematically cancels, skipped
  const float* bn1g_g = (const float*)d_in[6];
  const float* bn1g_b = (const float*)d_in[7];
  const float* W2g   = (const float*)d_in[8];
  const float* asrc2 = (const float*)d_in[9];
  const float* adst2 = (const float*)d_in[10];
  const float* b2g   = (const float*)d_in[11];
  const float* Wl1   = (const float*)d_in[12];
  const float* bl1   = (const float*)d_in[13];
  const float* Wr1   = (const float*)d_in[14];
  const float* bn1s_g = (const float*)d_in[15];
  const float* bn1s_b = (const float*)d_in[16];
  const float* Wl2   = (const float*)d_in[17];
  const float* bl2   = (const float*)d_in[18];
  const float* Wr2   = (const float*)d_in[19];
  const float* Wf1   = (const float*)d_in[20];
  const float* bf1   = (const float*)d_in[21];
  const float* Wf2   = (const float*)d_in[22];
  const float* bf2   = (const float*)d_in[23];
  float* out = (float*)d_out;

  const int n  = in_sizes[0] / 128;   // 100000 (divisible by 16)
  const int E  = in_sizes[1] / 2;     // 1600000
  const long EL = (long)E + n;        // with self loops
  const int H = 4, C = 64, HC = 256;
  const int* src = ei;
  const int* dst = ei + E;

  // ---- workspace carve-up (floats), 64-elem aligned ----
  float* ws = (float*)d_ws;
  long o = 0;
  auto take = [&](long cnt) { float* p = ws + o; o += (cnt + 63) & ~63L; return p; };
  float* h1    = take((long)n * HC);     // reused after GAT1 agg: agg1|s1|agg2
  float* g1    = take((long)n * HC);     // reused for fusion hidden at the end
  float* h2    = take((long)n * C);
  float* fused = take((long)n * 128);    // [g2 | s2]
  float* exbuf = take(EL * H);
  float* es1 = take((long)n * H); float* ed1 = take((long)n * H);
  float* m1  = take((long)n * H); float* den1 = take((long)n * H);
  float* es2 = take(n); float* ed2 = take(n);
  float* m2  = take(n); float* den2 = take(n);
  float* cnt = take(n);
  float* bnsum = take(256); float* bnsq = take(256);
  float* mu1 = take(256); float* rs1 = take(256);
  float* mu2 = take(64);  float* rs2 = take(64);
  // aliases into retired h1 region:
  float* agg1 = h1;                      // [n,128]
  float* s1   = h1 + (long)n * 128;      // [n,64]
  float* agg2 = h1 + (long)n * 192;      // [n,64]
  float* hfuse = g1;                     // [n,64]

  const dim3 B1(TPB);
  const int BN_BLOCKS = 512;

  // ===== GAT layer 1 =====
  k_wmma_gemm<0, false><<<dim3(n / 16, HC / 64), dim3(32), 0, stream>>>(
      x, 128, W1g, HC, nullptr, h1, HC, 128);
  k_gat_edot<<<cdiv((long)n * H * 32, TPB), B1, 0, stream>>>(h1, asrc1, adst1,
                                                             es1, ed1, n, H, C);
  k_fill<<<cdiv((long)n * H, TPB), B1, 0, stream>>>(m1, -1e30f, (long)n * H);
  k_fill<<<cdiv((long)n * H, TPB), B1, 0, stream>>>(den1, 0.f, (long)n * H);
  k_gat_max<<<cdiv(EL, TPB), B1, 0, stream>>>(src, dst, E, n, es1, ed1, m1, H);
  k_gat_expsum<<<cdiv(EL, TPB), B1, 0, stream>>>(src, dst, E, n, es1, ed1, m1, exbuf, den1, H);
  k_fill<<<cdiv((long)n * HC, TPB), B1, 0, stream>>>(g1, 0.f, (long)n * HC);
  k_gat_agg<<<cdiv(EL * H * 32, TPB), B1, 0, stream>>>(src, dst, E, n, h1, exbuf, den1,
                                                       g1, H, C, HC);
  // BN + ELU (in place)
  k_fill<<<1, TPB, 0, stream>>>(bnsum, 0.f, 256);
  k_fill<<<1, TPB, 0, stream>>>(bnsq, 0.f, 256);
  k_bn_part<<<BN_BLOCKS, HC, 0, stream>>>(g1, n, HC, bnsum, bnsq);
  k_bn_finish<<<1, HC, 0, stream>>>(bnsum, bnsq, n, HC, mu1, rs1);
  k_bn_apply<<<cdiv((long)n * HC, TPB), B1, 0, stream>>>(g1, (long)n * HC, HC,
                                                         mu1, rs1, bn1g_g, bn1g_b, 2);

  // ===== GAT layer 2 (heads=1) =====
  k_wmma_gemm<0, false><<<dim3(n / 16, C / 64), dim3(32), 0, stream>>>(
      g1, HC, W2g, C, nullptr, h2, C, HC);
  k_gat_edot<<<cdiv((long)n * 32, TPB), B1, 0, stream>>>(h2, asrc2, adst2, es2, ed2, n, 1, C);
  k_fill<<<cdiv((long)n, TPB), B1, 0, stream>>>(m2, -1e30f, n);
  k_fill<<<cdiv((long)n, TPB), B1, 0, stream>>>(den2, 0.f, n);
  k_gat_max<<<cdiv(EL, TPB), B1, 0, stream>>>(src, dst, E, n, es2, ed2, m2, 1);
  k_gat_expsum<<<cdiv(EL, TPB), B1, 0, stream>>>(src, dst, E, n, es2, ed2, m2, exbuf, den2, 1);
  k_fill<<<cdiv((long)n * 128, TPB), B1, 0, stream>>>(fused, 0.f, (long)n * 128);
  k_gat_agg<<<cdiv(EL * 32, TPB), B1, 0, stream>>>(src, dst, E, n, h2, exbuf, den2,
                                                   fused, 1, C, 128);
  k_bias_add<<<cdiv((long)n * C, TPB), B1, 0, stream>>>(fused, b2g, n, C, 128);

  // ===== SAGE branch (h1 region now free) =====
  k_fill<<<cdiv((long)n * 128, TPB), B1, 0, stream>>>(agg1, 0.f, (long)n * 128);
  k_fill<<<cdiv((long)n, TPB), B1, 0, stream>>>(cnt, 0.f, n);
  k_sage_agg<<<cdiv((long)E * 32, TPB), B1, 0, stream>>>(x, src, dst, E, agg1, cnt, 128);
  k_sage_div<<<cdiv((long)n * 128, TPB), B1, 0, stream>>>(agg1, cnt, n, 128);
  k_wmma_gemm<0, false><<<dim3(n / 16, C / 64), dim3(32), 0, stream>>>(
      x, 128, Wr1, C, nullptr, s1, C, 128);                          // s1 = x@Wr1
  k_wmma_gemm<1, true><<<dim3(n / 16, C / 64), dim3(32), 0, stream>>>(
      agg1, 128, Wl1, C, bl1, s1, C, 128);                           // s1 += agg1@Wl1+bl1
  k_fill<<<1, TPB, 0, stream>>>(bnsum, 0.f, 256);
  k_fill<<<1, TPB, 0, stream>>>(bnsq, 0.f, 256);
  k_bn_part<<<BN_BLOCKS, C, 0, stream>>>(s1, n, C, bnsum, bnsq);
  k_bn_finish<<<1, C, 0, stream>>>(bnsum, bnsq, n, C, mu2, rs2);
  k_bn_apply<<<cdiv((long)n * C, TPB), B1, 0, stream>>>(s1, (long)n * C, C,
                                                        mu2, rs2, bn1s_g, bn1s_b, 1);
  k_fill<<<cdiv((long)n * C, TPB), B1, 0, stream>>>(agg2, 0.f, (long)n * C);
  k_fill<<<cdiv((long)n, TPB), B1, 0, stream>>>(cnt, 0.f, n);
  k_sage_agg<<<cdiv((long)E * 32, TPB), B1, 0, stream>>>(s1, src, dst, E, agg2, cnt, C);
  k_sage_div<<<cdiv((long)n * C, TPB), B1, 0, stream>>>(agg2, cnt, n, C);
  k_wmma_gemm<0, false><<<dim3(n / 16, C / 64), dim3(32), 0, stream>>>(
      s1, C, Wr2, C, nullptr, fused + C, 128, C);                    // s2 = s1@Wr2
  k_wmma_gemm<1, true><<<dim3(n / 16, C / 64), dim3(32), 0, stream>>>(
      agg2, C, Wl2, C, bl2, fused + C, 128, C);                      // s2 += agg2@Wl2+bl2

  // ===== fusion MLP =====
  k_wmma_gemm<2, true><<<dim3(n / 16, C / 64), dim3(32), 0, stream>>>(
      fused, 128, Wf1, C, bf1, hfuse, C, 128);                       // relu(fused@Wf1+bf1)
  k_final<<<cdiv((long)n, TPB), B1, 0, stream>>>(hfuse, Wf2, bf2, out, n);
}